// DecouplingLoss_26671746908623
// MI455X (gfx1250) — compile-verified
//
#include <hip/hip_runtime.h>
#include <math.h>

// DecouplingLoss for MI455X (gfx1250, wave32).
// sh, sp: [8192, 2048] fp32. Chunks of 4 rows; per chunk:
//   sim = norm(sh) @ norm(sp)^T  -> -log(1-clip(sim)) mean
//   gram = norm(sh) @ norm(sh)^T -> pairwise dist -> row max -> hinge
// Strategy: 16 rows (4 chunks) per block; V_WMMA_F32_16X16X4_F32 accumulates
// raw sh.sp^T, sh.sh^T, sp.sp^T over K=2048 (8 waves x 256 K-slice each).
// Self-Gram diagonals give row norms, so normalization is a cheap epilogue.

typedef __attribute__((ext_vector_type(2))) float v2f;
typedef __attribute__((ext_vector_type(8))) float v8f;

#define DIMD     2048
#define ROWS     16          // 4 chunks of 4 views per block
#define WAVES    8
#define KSLICE   (DIMD / WAVES)   // 256

// 16x16 f32 C/D WMMA layout -> flat LDS index.
// lanes 0-15: M = vgpr,   N = lane
// lanes16-31: M = vgpr+8, N = lane-16
__device__ __forceinline__ int mat_idx(int m, int n) {
    return (m & 7) * 32 + ((m >> 3) << 4) + n;
}

__global__ __launch_bounds__(256) void decoupling_main(
    const float* __restrict__ sh, const float* __restrict__ sp,
    float* __restrict__ partials)
{
    __shared__ float red[WAVES * 3 * 256];   // 24 KB of the 320 KB WGP LDS
    __shared__ float chunk_part[4];

    const int tid   = threadIdx.x;
    const int lane  = tid & 31;
    const int wave  = tid >> 5;
    const int row   = lane & 15;
    const int khalf = (lane >> 4) << 1;      // 0 for lanes 0-15, 2 for 16-31

    const size_t rowbase = (size_t)(blockIdx.x * ROWS + row) * DIMD;
    const float* pa = sh + rowbase + (size_t)wave * KSLICE + khalf;
    const float* pb = sp + rowbase + (size_t)wave * KSLICE + khalf;

    v8f accSim = {};   // sh . sp^T   (raw)
    v8f accShG = {};   // sh . sh^T   (raw; diag = ||sh row||^2)
    v8f accSpG = {};   // sp . sp^T   (raw; diag = ||sp row||^2)

    #pragma unroll 8
    for (int k = 0; k < KSLICE; k += 4) {
        v2f a = *(const v2f*)(pa + k);
        v2f b = *(const v2f*)(pb + k);
        accSim = __builtin_amdgcn_wmma_f32_16x16x4_f32(
            false, a, false, b, (short)0, accSim, false, false);
        accShG = __builtin_amdgcn_wmma_f32_16x16x4_f32(
            false, a, false, a, (short)0, accShG, false, false);
        accSpG = __builtin_amdgcn_wmma_f32_16x16x4_f32(
            false, b, false, b, (short)0, accSpG, false, false);
    }

    // Stash per-wave 16x16 partial tiles.
    float* base = red + wave * (3 * 256);
    #pragma unroll
    for (int v = 0; v < 8; ++v) {
        base[0 * 256 + v * 32 + lane] = accSim[v];
        base[1 * 256 + v * 32 + lane] = accShG[v];
        base[2 * 256 + v * 32 + lane] = accSpG[v];
    }
    __syncthreads();

    // Cross-wave reduction: thread t owns matrix entry t of each tile.
    float s0 = 0.f, s1 = 0.f, s2 = 0.f;
    #pragma unroll
    for (int w = 0; w < WAVES; ++w) {
        const float* b2 = red + w * (3 * 256);
        s0 += b2[0 * 256 + tid];
        s1 += b2[1 * 256 + tid];
        s2 += b2[2 * 256 + tid];
    }
    __syncthreads();
    red[0 * 256 + tid] = s0;
    red[1 * 256 + tid] = s1;
    red[2 * 256 + tid] = s2;
    __syncthreads();

    // Epilogue: one thread per 4-row chunk.
    if (tid < 4) {
        const float* SIM = red;
        const float* SHG = red + 256;
        const float* SPG = red + 512;
        const int m0 = tid * 4;

        float invsh[4], invsp[4], gdiag[4];
        #pragma unroll
        for (int i = 0; i < 4; ++i) {
            int m = m0 + i;
            float nsh = sqrtf(SHG[mat_idx(m, m)]);
            float nsp = sqrtf(SPG[mat_idx(m, m)]);
            invsh[i] = 1.0f / fmaxf(nsh, 1e-12f);
            invsp[i] = 1.0f / fmaxf(nsp, 1e-12f);
        }
        #pragma unroll
        for (int i = 0; i < 4; ++i)
            gdiag[i] = SHG[mat_idx(m0 + i, m0 + i)] * invsh[i] * invsh[i];

        float special = 0.f, shsh = 0.f;
        #pragma unroll
        for (int i = 0; i < 4; ++i) {
            float rowmax = -INFINITY;
            #pragma unroll
            for (int j = 0; j < 4; ++j) {
                float s = SIM[mat_idx(m0 + i, m0 + j)] * invsh[i] * invsp[j];
                s = fminf(fmaxf(s, 0.0005f), 0.9995f);
                special += -logf(1.0f - s);
                if (j != i) {
                    float g  = SHG[mat_idx(m0 + i, m0 + j)] * invsh[i] * invsh[j];
                    float d2 = fmaxf(gdiag[i] + gdiag[j] - 2.0f * g, 1e-12f);
                    rowmax = fmaxf(rowmax, sqrtf(d2));
                }
            }
            shsh += fmaxf(rowmax + 0.3f, 0.0f);
        }
        chunk_part[tid] = special * (1.0f / 16.0f) + 0.01f * (shsh * 0.25f);
    }
    __syncthreads();
    if (tid == 0)
        partials[blockIdx.x] =
            chunk_part[0] + chunk_part[1] + chunk_part[2] + chunk_part[3];
}

// Deterministic fixed-order final reduction (no float atomics).
__global__ __launch_bounds__(256) void decoupling_reduce(
    const float* __restrict__ partials, float* __restrict__ out, int n)
{
    __shared__ float s[256];
    const int t = threadIdx.x;
    float v = 0.f;
    for (int i = t; i < n; i += 256) v += partials[i];
    s[t] = v;
    __syncthreads();
    #pragma unroll
    for (int stride = 128; stride > 0; stride >>= 1) {
        if (t < stride) s[t] += s[t + stride];
        __syncthreads();
    }
    if (t == 0) out[0] = s[0];
}

extern "C" void kernel_launch(void* const* d_in, const int* in_sizes, int n_in,
                              void* d_out, int out_size, void* d_ws, size_t ws_size,
                              hipStream_t stream) {
    const float* sh = (const float*)d_in[0];
    const float* sp = (const float*)d_in[1];
    float* out = (float*)d_out;
    float* ws  = (float*)d_ws;

    const int B = in_sizes[0] / DIMD;     // 8192
    const int nblocks = B / ROWS;         // 512

    decoupling_main<<<nblocks, 256, 0, stream>>>(sh, sp, ws);
    decoupling_reduce<<<1, 256, 0, stream>>>(ws, out, nblocks);
}